// TrainableRetinex_18081812316573
// MI455X (gfx1250) — compile-verified
//
#include <hip/hip_runtime.h>

// TrainableRetinex on MI455X (gfx1250):
// separable 15x15 Gaussian blur expressed as banded-Toeplitz GEMMs on
// V_WMMA_F32_16X16X4_F32, staged through LDS; retinex + alpha + clamp +
// 4x channel-repeat fused into one bandwidth-bound pass.

typedef __attribute__((ext_vector_type(2))) float v2f;
typedef __attribute__((ext_vector_type(8))) float v8f;

#define TILE      16
#define HALO      7
#define XT_STRIDE 33   // 32 used cols + 1 pad (bank-conflict avoidance)
#define HB_STRIDE 17   // 16 used cols + 1 pad

// Normalized 1D gaussian weight g[i] for i in [0,14], 0 outside.
// sigma=5 -> exp(-d^2/50); 1/sum(exp) = 0.09202463 (S = 10.8666566)
__device__ __forceinline__ float gaussw(int i) {
  float d = (float)(i - 7);
  float e = __expf(d * d * -0.02f);
  return ((unsigned)i <= 14u) ? e * 0.09202463f : 0.0f;
}

__global__ __launch_bounds__(32)
void retinex_wmma_kernel(const float* __restrict__ x,
                         const float* __restrict__ alpha,
                         float* __restrict__ out) {
  __shared__ float lds_x[32 * XT_STRIDE];   // input tile + halo, zero padded
  __shared__ float lds_hb[32 * HB_STRIDE];  // horizontally blurred rows

  const int lane = threadIdx.x;   // 0..31 (wave32)
  const int half = lane >> 4;     // 0: K=0,1 / 1: K=2,3 (WMMA f32 layout)
  const int l15  = lane & 15;

  const int bid   = blockIdx.x;
  const int tw    = bid & 15;
  const int th    = (bid >> 4) & 15;
  const int plane = bid >> 8;      // == b*64 + c  (NCHW flat plane index)
  const int c     = plane & 63;
  const int b     = plane >> 6;

  const int h0 = th * TILE;
  const int w0 = tw * TILE;

  const float* xp = x + (size_t)plane * 65536;  // 256*256 plane

  // ---- load 32x32 tile covering rows/cols [h0-7, h0+23), zero padded ----
  {
    const int  gw  = w0 - HALO + lane;          // lane = tile column
    const bool wok = (lane < 30) && (gw >= 0) && (gw < 256);
#pragma unroll
    for (int r = 0; r < 32; ++r) {
      const int gh = h0 - HALO + r;
      float v = 0.0f;
      if (wok && (r < 30) && (gh >= 0) && (gh < 256))
        v = xp[gh * 256 + gw];
      lds_x[r * XT_STRIDE + lane] = v;
    }
  }
  __syncthreads();

  // ---- horizontal pass: hb(r,w) = sum_k Xtile(r,k) * g[k-w] ----
  // Two 16-row groups cover the 30 (padded to 32) halo rows.
#pragma unroll
  for (int grp = 0; grp < 2; ++grp) {
    v8f acc = {};
#pragma unroll
    for (int s = 0; s < 8; ++s) {          // K = 32 in steps of 4
      const int k0 = 4 * s + 2 * half;
      v2f a, bm;
      a.x  = lds_x[(grp * 16 + l15) * XT_STRIDE + k0];      // A(M=l15, k0)
      a.y  = lds_x[(grp * 16 + l15) * XT_STRIDE + k0 + 1];
      bm.x = gaussw(k0 - l15);                              // B(k0, N=l15)
      bm.y = gaussw(k0 + 1 - l15);
      acc = __builtin_amdgcn_wmma_f32_16x16x4_f32(false, a, false, bm,
                                                  (short)0, acc, false, false);
    }
#pragma unroll
    for (int j = 0; j < 8; ++j)            // C/D layout: M = j + 8*half
      lds_hb[(grp * 16 + j + 8 * half) * HB_STRIDE + l15] = acc[j];
  }
  __syncthreads();

  // ---- vertical pass: out(h,w) = sum_k g[k-h] * hb(k,w) ----
  v8f vacc = {};
#pragma unroll
  for (int s = 0; s < 8; ++s) {
    const int k0 = 4 * s + 2 * half;
    v2f a, bm;
    a.x  = gaussw(k0 - l15);                                // A(M=l15, k0)
    a.y  = gaussw(k0 + 1 - l15);
    bm.x = lds_hb[k0 * HB_STRIDE + l15];                    // B(k0, N=l15)
    bm.y = lds_hb[(k0 + 1) * HB_STRIDE + l15];
    vacc = __builtin_amdgcn_wmma_f32_16x16x4_f32(false, a, false, bm,
                                                 (short)0, vacc, false, false);
  }

  // ---- retinex + alpha scale + clamp; write the 4 repeated channels ----
  const float  ac    = alpha[c];
  const size_t obase = (size_t)(b * 256 + c * 4) * 65536;
#pragma unroll
  for (int j = 0; j < 8; ++j) {
    const int   M    = j + 8 * half;
    const int   h    = h0 + M;
    const int   w    = w0 + l15;
    const float xv   = lds_x[(M + HALO) * XT_STRIDE + (l15 + HALO)];
    const float blur = vacc[j];
    float en = ac * (logf(xv + 1e-6f) - logf(blur + 1e-6f));
    en = fminf(2.0f, fmaxf(-2.0f, en));
    const size_t o = obase + (size_t)h * 256 + w;
    out[o]              = en;
    out[o + 65536]      = en;
    out[o + 2 * 65536]  = en;
    out[o + 3 * 65536]  = en;
  }
}

extern "C" void kernel_launch(void* const* d_in, const int* in_sizes, int n_in,
                              void* d_out, int out_size, void* d_ws, size_t ws_size,
                              hipStream_t stream) {
  (void)in_sizes; (void)n_in; (void)out_size; (void)d_ws; (void)ws_size;
  const float* x     = (const float*)d_in[0];
  const float* alpha = (const float*)d_in[1];
  float*       out   = (float*)d_out;

  // 16x16 tiles: 16*16 tiles/plane * 256 planes, one wave32 per tile.
  dim3 grid(65536), block(32);
  hipLaunchKernelGGL(retinex_wmma_kernel, grid, block, 0, stream, x, alpha, out);
}